// WeightedAggEdge_37838661878216
// MI455X (gfx1250) — compile-verified
//
#include <hip/hip_runtime.h>

// ---------------------------------------------------------------------------
// Sizes (match reference file constants)
// ---------------------------------------------------------------------------
static constexpr int N_NODES  = 50000;
static constexpr int N_EDGES  = 800000;
static constexpr int NODE_IN  = 128;
static constexpr int NODE_OUT = 64;
static constexpr int EDGE_IN  = 64;
static constexpr int EDGE_OUT = 64;
static constexpr float NEG_SLOPE = 0.01f;

typedef __attribute__((ext_vector_type(2))) float v2f;
typedef __attribute__((ext_vector_type(8))) float v8f;

// ---------------------------------------------------------------------------
// WMMA f32 strip core:  acc[16 x 64] = X[r0:r0+16, :K] @ W[64 x K]^T
// One wave, 4 accumulator tiles of V_WMMA_F32_16X16X4_F32, K stepped by 4.
// A frag (16x4 f32): lane (hi,lo) holds X[r0+lo][k0+2hi .. +1]  (float2)
// B frag (4x16 f32): lane holds W[16t+lo][k0+2hi .. +1]          (float2)
// C/D tile: VGPR j -> (M=j+8*hi, N=lo)
// ---------------------------------------------------------------------------
template <int K>
__device__ __forceinline__ void wmma_strip(const float* __restrict__ X,
                                           const float* __restrict__ W,
                                           int r0, int lo, int hi,
                                           v8f& c0, v8f& c1, v8f& c2, v8f& c3) {
  const int kb = hi * 2;
  const float* xrow = X + (size_t)(r0 + lo) * K + kb;
  const float* w0 = W + (size_t)(lo +  0) * K + kb;
  const float* w1 = W + (size_t)(lo + 16) * K + kb;
  const float* w2 = W + (size_t)(lo + 32) * K + kb;
  const float* w3 = W + (size_t)(lo + 48) * K + kb;
  for (int k0 = 0; k0 < K; k0 += 4) {
    v2f a  = *(const v2f*)(xrow + k0);
    v2f b0 = *(const v2f*)(w0 + k0);
    v2f b1 = *(const v2f*)(w1 + k0);
    v2f b2 = *(const v2f*)(w2 + k0);
    v2f b3 = *(const v2f*)(w3 + k0);
    c0 = __builtin_amdgcn_wmma_f32_16x16x4_f32(false, a, false, b0, (short)0, c0, false, false);
    c1 = __builtin_amdgcn_wmma_f32_16x16x4_f32(false, a, false, b1, (short)0, c1, false, false);
    c2 = __builtin_amdgcn_wmma_f32_16x16x4_f32(false, a, false, b2, (short)0, c2, false, false);
    c3 = __builtin_amdgcn_wmma_f32_16x16x4_f32(false, a, false, b3, (short)0, c3, false, false);
  }
}

__device__ __forceinline__ void store_strip(float* __restrict__ out, int r0, int lo, int hi,
                                            const v8f& c0, const v8f& c1,
                                            const v8f& c2, const v8f& c3) {
  float* orow = out + (size_t)(r0 + 8 * hi) * 64 + lo;
#pragma unroll
  for (int j = 0; j < 8; ++j) {
    float* p = orow + (size_t)j * 64;
    p[0]  = c0[j];
    p[16] = c1[j];
    p[32] = c2[j];
    p[48] = c3[j];
  }
}

// Butterfly sum over the 16 lanes of this half-wave (masks < 16 never cross
// the lane-16 boundary on wave32).
__device__ __forceinline__ float half_reduce(float p) {
  p += __shfl_xor(p, 1, 32);
  p += __shfl_xor(p, 2, 32);
  p += __shfl_xor(p, 4, 32);
  p += __shfl_xor(p, 8, 32);
  return p;
}

// ---------------------------------------------------------------------------
// Monotonic float <-> uint key (total order preserved) for atomic segment max
// ---------------------------------------------------------------------------
__device__ __forceinline__ unsigned fkey(float f) {
  unsigned u = __float_as_uint(f);
  return (u & 0x80000000u) ? ~u : (u | 0x80000000u);
}
__device__ __forceinline__ float funkey(unsigned k) {
  unsigned u = (k & 0x80000000u) ? (k ^ 0x80000000u) : ~k;
  return __uint_as_float(u);
}

// ---------------------------------------------------------------------------
// Kernel 1: node_feat = h @ Wn^T, fused with
//   s_n[r] = dot(node_feat[r], Wa[64:128]),  mkey[r]=0,  denom[r]=0
// ---------------------------------------------------------------------------
__global__ __launch_bounds__(256) void gemm_node_fused(const float* __restrict__ X,
                                                       const float* __restrict__ W,
                                                       const float* __restrict__ Wa,
                                                       float* __restrict__ out,
                                                       float* __restrict__ s_n,
                                                       unsigned* __restrict__ mkey,
                                                       float* __restrict__ denom,
                                                       int nstrips) {
  const int wave = blockIdx.x * (blockDim.x >> 5) + (threadIdx.x >> 5);
  if (wave >= nstrips) return;               // wave-uniform: EXEC stays full
  const int lane = threadIdx.x & 31;
  const int lo = lane & 15;
  const int hi = lane >> 4;
  const int r0 = wave * 16;

  v8f c0 = {}, c1 = {}, c2 = {}, c3 = {};
  wmma_strip<NODE_IN>(X, W, r0, lo, hi, c0, c1, c2, c3);
  store_strip(out, r0, lo, hi, c0, c1, c2, c3);

  // Row-dot with node half of Wa while C is in registers.
  const float wa0 = Wa[NODE_OUT + lo];
  const float wa1 = Wa[NODE_OUT + 16 + lo];
  const float wa2 = Wa[NODE_OUT + 32 + lo];
  const float wa3 = Wa[NODE_OUT + 48 + lo];
#pragma unroll
  for (int j = 0; j < 8; ++j) {
    float p = c0[j] * wa0 + c1[j] * wa1 + c2[j] * wa2 + c3[j] * wa3;
    p = half_reduce(p);                      // row sum, all lanes in half hold it
    if (lo == j) {                           // lane j (hi=0) and lane 16+j (hi=1)
      int r = r0 + 8 * hi + j;
      s_n[r] = p;
      mkey[r] = 0u;                          // key-space minimum
      denom[r] = 0.f;
    }
  }
}

// ---------------------------------------------------------------------------
// Kernel 2: e_w = e @ We^T, fused with
//   v = leaky_relu(dot(e_w[r], Wa[0:64]) + s_n[src[r]])
//   elr[r] = v ; atomicMax(mkey[src[r]], key(v))
// ---------------------------------------------------------------------------
__global__ __launch_bounds__(256) void gemm_edge_fused(const float* __restrict__ X,
                                                       const float* __restrict__ W,
                                                       const float* __restrict__ Wa,
                                                       const float* __restrict__ s_n,
                                                       const int* __restrict__ src,
                                                       float* __restrict__ out,
                                                       float* __restrict__ elr,
                                                       unsigned* __restrict__ mkey,
                                                       int nstrips) {
  const int wave = blockIdx.x * (blockDim.x >> 5) + (threadIdx.x >> 5);
  if (wave >= nstrips) return;
  const int lane = threadIdx.x & 31;
  const int lo = lane & 15;
  const int hi = lane >> 4;
  const int r0 = wave * 16;

  v8f c0 = {}, c1 = {}, c2 = {}, c3 = {};
  wmma_strip<EDGE_IN>(X, W, r0, lo, hi, c0, c1, c2, c3);
  store_strip(out, r0, lo, hi, c0, c1, c2, c3);

  // Row-dot with edge half of Wa while C is in registers.
  const float wa0 = Wa[lo];
  const float wa1 = Wa[16 + lo];
  const float wa2 = Wa[32 + lo];
  const float wa3 = Wa[48 + lo];
#pragma unroll
  for (int j = 0; j < 8; ++j) {
    float p = c0[j] * wa0 + c1[j] * wa1 + c2[j] * wa2 + c3[j] * wa3;
    p = half_reduce(p);
    if (lo == j) {
      int r = r0 + 8 * hi + j;
      int s = src[r];
      float a = p + s_n[s];
      float v = (a >= 0.f) ? a : NEG_SLOPE * a;
      elr[r] = v;
      atomicMax(mkey + s, fkey(v));
    }
  }
}

// Per-edge: ex = exp(v - m[src]); segment sum into denom.
__global__ void edge_exp(const int* __restrict__ src, const unsigned* __restrict__ mkey,
                         float* __restrict__ elr, float* __restrict__ denom, int n) {
  int i = blockIdx.x * blockDim.x + threadIdx.x;
  if (i >= n) return;
  int s = src[i];
  float ex = __expf(elr[i] - funkey(mkey[s]));
  elr[i] = ex;
  atomicAdd(denom + s, ex);
}

// Per (edge, float4-chunk): e_weighted = gamma * e_w, in place in d_out.
__global__ void edge_scale(float* __restrict__ ew, const float* __restrict__ elr,
                           const float* __restrict__ denom, const int* __restrict__ src,
                           int nchunks) {
  int gid = blockIdx.x * blockDim.x + threadIdx.x;
  if (gid >= nchunks) return;
  int e = gid >> 4;          // 16 float4 chunks per 64-float row
  int c = gid & 15;
  float gamma = elr[e] / denom[src[e]];
  float4* p = (float4*)(ew + (size_t)e * EDGE_OUT + c * 4);
  float4 v = *p;
  v.x *= gamma; v.y *= gamma; v.z *= gamma; v.w *= gamma;
  *p = v;
}

// ---------------------------------------------------------------------------
// Launch
// ---------------------------------------------------------------------------
extern "C" void kernel_launch(void* const* d_in, const int* in_sizes, int n_in,
                              void* d_out, int out_size, void* d_ws, size_t ws_size,
                              hipStream_t stream) {
  const float* h  = (const float*)d_in[0];   // [N_NODES, 128]
  const float* e  = (const float*)d_in[1];   // [N_EDGES, 64]
  const float* Wn = (const float*)d_in[2];   // [64, 128]
  const float* We = (const float*)d_in[3];   // [64, 64]
  const float* Wa = (const float*)d_in[4];   // [1, 128]
  const int*  src = (const int*)d_in[5];     // [N_EDGES]

  float* nf = (float*)d_out;                              // node_feat
  float* ew = (float*)d_out + (size_t)N_NODES * NODE_OUT; // e_w -> e_weighted

  float*    s_n   = (float*)d_ws;
  unsigned* mkey  = (unsigned*)(s_n + N_NODES);
  float*    denom = (float*)(mkey + N_NODES);
  float*    elr   = denom + N_NODES;         // per-edge logits / exps (3.2 MB)

  // 1) node GEMM + s_n + segment init  (3125 strips of 16 rows, 8 waves/block)
  {
    int nstrips = N_NODES / 16;
    int blocks = (nstrips + 7) / 8;
    gemm_node_fused<<<blocks, 256, 0, stream>>>(h, Wn, Wa, nf, s_n, mkey, denom, nstrips);
  }
  // 2) edge GEMM + logit + leaky-relu + segment max  (50000 strips)
  {
    int nstrips = N_EDGES / 16;
    int blocks = (nstrips + 7) / 8;
    gemm_edge_fused<<<blocks, 256, 0, stream>>>(e, We, Wa, s_n, src, ew, elr, mkey, nstrips);
  }
  // 3) exp + segment sum
  edge_exp<<<(N_EDGES + 255) / 256, 256, 0, stream>>>(src, mkey, elr, denom, N_EDGES);
  // 4) scale e_w in place -> e_weighted
  {
    int nchunks = N_EDGES * 16;
    edge_scale<<<(nchunks + 255) / 256, 256, 0, stream>>>(ew, elr, denom, src, nchunks);
  }
}